// DualEmbedding_25872882991624
// MI455X (gfx1250) — compile-verified
//
#include <hip/hip_runtime.h>
#include <hip/hip_bf16.h>

typedef __attribute__((ext_vector_type(2))) float v2f;
typedef __attribute__((ext_vector_type(4))) float v4f;
typedef __attribute__((ext_vector_type(8))) float v8f;

#define N_NODES 100000
#define N_EDGES 1000000
#define NODE_IN 128
#define EDGE_IN 64
#define D_MODEL 128
#define LDS_STRIDE 132   // 132 % 64 == 4 -> row halves land 32 banks apart, conflict-free

__device__ __forceinline__ float wave_reduce_sum(float v) {
    // wave32 butterfly reduction
    #pragma unroll
    for (int m = 16; m >= 1; m >>= 1)
        v += __shfl_xor(v, m, 32);
    return v;
}

// ---------------------------------------------------------------------------
// Kernel 1: h_node[N,128] = x_node[N,128] @ W_node[128,128]^T + b_node
// grid = N/16 blocks * 256 thr (8 waves); wave w -> 16x16 tile, cols 16w..16w+15
// ---------------------------------------------------------------------------
__global__ __launch_bounds__(256)
void node_proj_kernel(const float* __restrict__ x_node,
                      const float* __restrict__ W_node,
                      const float* __restrict__ b_node,
                      float* __restrict__ h_node) {
    const int wave = threadIdx.x >> 5;
    const int lane = threadIdx.x & 31;
    const int tl   = lane & 15;       // position within half-wave
    const int hi   = lane >> 4;       // half-wave select (K offset +2 / M offset +8)
    const int r0   = blockIdx.x * 16; // output row tile
    const int n0   = wave * 16;       // output col tile

    const float* arow = x_node + (size_t)(r0 + tl) * NODE_IN + 2 * hi;
    const float* brow = W_node + (size_t)(n0 + tl) * NODE_IN + 2 * hi; // B[k][n] = W[n][k]

    v8f acc0 = {};
    v8f acc1 = {};
    #pragma unroll
    for (int k = 0; k < NODE_IN; k += 8) {
        v2f a0 = *(const v2f*)(arow + k);
        v2f b0 = *(const v2f*)(brow + k);
        acc0 = __builtin_amdgcn_wmma_f32_16x16x4_f32(false, a0, false, b0,
                                                     (short)0, acc0, false, false);
        v2f a1 = *(const v2f*)(arow + k + 4);
        v2f b1 = *(const v2f*)(brow + k + 4);
        acc1 = __builtin_amdgcn_wmma_f32_16x16x4_f32(false, a1, false, b1,
                                                     (short)0, acc1, false, false);
    }

    const float bias = b_node[n0 + tl];
    float* orow = h_node + (size_t)(r0 + 8 * hi) * D_MODEL + n0 + tl;
    #pragma unroll
    for (int r = 0; r < 8; ++r)
        orow[(size_t)r * D_MODEL] = acc0[r] + acc1[r] + bias;   // row = r0 + r + 8*hi
}

// ---------------------------------------------------------------------------
// Kernel 2 (fused): per block of 16 edges
//   tile = x_edge[16x64] @ W_edge[128x64]^T           (8 waves x one 16x16 WMMA tile)
//   v    = tile + b_edge + 0.5*(h_node[src]+h_node[dst])
//   out  = LayerNorm(v) * gamma + beta                (wave w owns rows 2w, 2w+1)
// ---------------------------------------------------------------------------
__global__ __launch_bounds__(256)
void edge_fused_kernel(const float* __restrict__ x_edge,
                       const int*   __restrict__ src,
                       const int*   __restrict__ dst,
                       const float* __restrict__ W_edge,
                       const float* __restrict__ b_edge,
                       const float* __restrict__ h_node,
                       const float* __restrict__ ln_gamma,
                       const float* __restrict__ ln_beta,
                       float* __restrict__ out) {
    __shared__ float tile[16 * LDS_STRIDE];

    const int wave = threadIdx.x >> 5;
    const int lane = threadIdx.x & 31;
    const int tl   = lane & 15;
    const int hi   = lane >> 4;
    const size_t e0 = (size_t)blockIdx.x * 16;
    const int n0   = wave * 16;

    // ---- GEMM phase: 16x16 tile of x_edge @ W_edge^T, K = 64 ----
    const float* arow = x_edge + (e0 + tl) * EDGE_IN + 2 * hi;
    const float* brow = W_edge + (size_t)(n0 + tl) * EDGE_IN + 2 * hi;

    v8f acc0 = {};
    v8f acc1 = {};
    #pragma unroll
    for (int k = 0; k < EDGE_IN; k += 8) {
        v2f a0 = __builtin_nontemporal_load((const v2f*)(arow + k));      // streamed
        v2f b0 = *(const v2f*)(brow + k);                                 // L2-hot
        acc0 = __builtin_amdgcn_wmma_f32_16x16x4_f32(false, a0, false, b0,
                                                     (short)0, acc0, false, false);
        v2f a1 = __builtin_nontemporal_load((const v2f*)(arow + k + 4));
        v2f b1 = *(const v2f*)(brow + k + 4);
        acc1 = __builtin_amdgcn_wmma_f32_16x16x4_f32(false, a1, false, b1,
                                                     (short)0, acc1, false, false);
    }

    #pragma unroll
    for (int r = 0; r < 8; ++r)
        tile[(r + 8 * hi) * LDS_STRIDE + n0 + tl] = acc0[r] + acc1[r];

    __syncthreads();

    // ---- gather + bias + LayerNorm phase: wave w handles rows 2w, 2w+1 ----
    const v4f g4  = *(const v4f*)(ln_gamma + 4 * lane);
    const v4f bt4 = *(const v4f*)(ln_beta  + 4 * lane);
    const v4f be4 = *(const v4f*)(b_edge   + 4 * lane);

    #pragma unroll
    for (int rr = 0; rr < 2; ++rr) {
        const int r = 2 * wave + rr;
        const size_t e = e0 + r;
        const int s = src[e];
        const int d = dst[e];
        // coalesced 512B row gathers; h_node (51MB) stays resident in 192MB L2
        const v4f hs = *(const v4f*)(h_node + (size_t)s * D_MODEL + 4 * lane);
        const v4f hd = *(const v4f*)(h_node + (size_t)d * D_MODEL + 4 * lane);
        const v4f t  = *(const v4f*)(&tile[r * LDS_STRIDE + 4 * lane]);

        float vx = t.x + be4.x + 0.5f * (hs.x + hd.x);
        float vy = t.y + be4.y + 0.5f * (hs.y + hd.y);
        float vz = t.z + be4.z + 0.5f * (hs.z + hd.z);
        float vw = t.w + be4.w + 0.5f * (hs.w + hd.w);

        const float mu = wave_reduce_sum(vx + vy + vz + vw) * (1.0f / D_MODEL);
        const float dx = vx - mu, dy = vy - mu, dz = vz - mu, dw = vw - mu;
        const float var = wave_reduce_sum(dx * dx + dy * dy + dz * dz + dw * dw)
                          * (1.0f / D_MODEL);
        const float rstd = rsqrtf(var + 1e-5f);

        v4f o;
        o.x = dx * rstd * g4.x + bt4.x;
        o.y = dy * rstd * g4.y + bt4.y;
        o.z = dz * rstd * g4.z + bt4.z;
        o.w = dw * rstd * g4.w + bt4.w;
        __builtin_nontemporal_store(o, (v4f*)(out + e * D_MODEL + 4 * lane));
    }
}

extern "C" void kernel_launch(void* const* d_in, const int* in_sizes, int n_in,
                              void* d_out, int out_size, void* d_ws, size_t ws_size,
                              hipStream_t stream) {
    const float* x_node     = (const float*)d_in[0];
    const float* x_edge     = (const float*)d_in[1];
    const int*   edge_index = (const int*)d_in[2];   // [2, E]; integer inputs arrive as int32
    const float* W_edge     = (const float*)d_in[3];
    const float* b_edge     = (const float*)d_in[4];
    const float* W_node     = (const float*)d_in[5];
    const float* b_node     = (const float*)d_in[6];
    const float* ln_gamma   = (const float*)d_in[7];
    const float* ln_beta    = (const float*)d_in[8];
    float* out    = (float*)d_out;
    float* h_node = (float*)d_ws;                    // 100000*128*4 = 51.2 MB scratch

    node_proj_kernel<<<N_NODES / 16, 256, 0, stream>>>(x_node, W_node, b_node, h_node);

    edge_fused_kernel<<<N_EDGES / 16, 256, 0, stream>>>(
        x_edge, edge_index, edge_index + N_EDGES,
        W_edge, b_edge, h_node, ln_gamma, ln_beta, out);
}